// GT_Model_52690658787913
// MI455X (gfx1250) — compile-verified
//
#include <hip/hip_runtime.h>
#include <hip/hip_bf16.h>
#include <math.h>

// ---------------------------------------------------------------------------
// GT TransformerConv (3 layers, H=4) for MI455X / gfx1250 (wave32, WMMA)
// ---------------------------------------------------------------------------
#define NNODES 10000
#define NEDGES 320000
#define NHEADS 4

typedef float v2f __attribute__((ext_vector_type(2)));
typedef float v8f __attribute__((ext_vector_type(8)));

// ------------------------- helpers -----------------------------------------
__device__ __forceinline__ unsigned fkey(float f) {
  unsigned b = __float_as_uint(f);
  return (b & 0x80000000u) ? ~b : (b | 0x80000000u);
}
__device__ __forceinline__ float funkey(unsigned k) {
  return __uint_as_float((k & 0x80000000u) ? (k & 0x7fffffffu) : ~k);
}

// ------------------------- dense projection GEMM ----------------------------
// One wave computes a 16x64 tile of O = A(MxK) @ W(K x NW) + bias, fp32 WMMA.
// A-matrix 16x4 layout (ISA 7.12.2): lanes 0-15 rows M=0..15 hold K=+0/+1,
// lanes 16-31 hold K=+2/+3. C/D: vgpr r -> M = r + 8*(lane/16), N = lane%16.
// Ragged-N handling: column indices are CLAMPED (not predicated) for loads —
// an out-of-range column only feeds accumulator columns that are never stored
// (WMMA column n depends only on B column n), so the K-loop is branch-free.
template <int NW>
__global__ __launch_bounds__(32) void gemm_bias_wmma(
    const float* __restrict__ A, const float* __restrict__ W,
    const float* __restrict__ bias, float* __restrict__ O, int K) {
  const int lane = threadIdx.x & 31;
  const int half = lane >> 4;
  const int lm = lane & 15;
  const int row = blockIdx.x * 16 + lm;  // M = 10000 is a multiple of 16
  const int c0 = blockIdx.y * 64 + lm;   // lane's column in subtile 0
  // clamped per-lane column offsets (hoisted out of the K loop)
  const int cc0 = (c0      < NW) ? c0      : NW - 1;
  const int cc1 = (c0 + 16 < NW) ? c0 + 16 : NW - 1;
  const int cc2 = (c0 + 32 < NW) ? c0 + 32 : NW - 1;
  const int cc3 = (c0 + 48 < NW) ? c0 + 48 : NW - 1;

  const float* __restrict__ Ar = A + (size_t)row * K;

  v8f acc0 = {}, acc1 = {}, acc2 = {}, acc3 = {};
#pragma unroll 2
  for (int k0 = 0; k0 < K; k0 += 4) {
    const int ka = k0 + half * 2;
    const float* __restrict__ Wr0 = W + (size_t)ka * NW;
    const float* __restrict__ Wr1 = Wr0 + NW;
    v2f a, b0, b1, b2, b3;
    a.x = Ar[ka];
    a.y = Ar[ka + 1];
    b0.x = Wr0[cc0]; b0.y = Wr1[cc0];
    b1.x = Wr0[cc1]; b1.y = Wr1[cc1];
    b2.x = Wr0[cc2]; b2.y = Wr1[cc2];
    b3.x = Wr0[cc3]; b3.y = Wr1[cc3];
    acc0 = __builtin_amdgcn_wmma_f32_16x16x4_f32(false, a, false, b0, (short)0, acc0, false, false);
    acc1 = __builtin_amdgcn_wmma_f32_16x16x4_f32(false, a, false, b1, (short)0, acc1, false, false);
    acc2 = __builtin_amdgcn_wmma_f32_16x16x4_f32(false, a, false, b2, (short)0, acc2, false, false);
    acc3 = __builtin_amdgcn_wmma_f32_16x16x4_f32(false, a, false, b3, (short)0, acc3, false, false);
  }
  const float bb0 = bias[cc0], bb1 = bias[cc1], bb2 = bias[cc2], bb3 = bias[cc3];
#pragma unroll
  for (int r = 0; r < 8; ++r) {
    const int orow = blockIdx.x * 16 + r + half * 8;
    float* op = O + (size_t)orow * NW;
    if (c0      < NW) op[c0]      = acc0[r] + bb0;
    if (c0 + 16 < NW) op[c0 + 16] = acc1[r] + bb1;
    if (c0 + 32 < NW) op[c0 + 32] = acc2[r] + bb2;
    if (c0 + 48 < NW) op[c0 + 48] = acc3[r] + bb3;
  }
}

// ------------------------- CSR-by-destination build -------------------------
__global__ void deg_count_k(const int* __restrict__ dst, int* __restrict__ deg, int E) {
  int e = blockIdx.x * blockDim.x + threadIdx.x;
  if (e < E) atomicAdd(&deg[dst[e]], 1);
}

__global__ __launch_bounds__(1024) void scan_k(const int* __restrict__ deg,
                                               int* __restrict__ row_start, int n) {
  __shared__ int part[1024];
  const int t = threadIdx.x;
  const int CH = (n + 1023) / 1024;
  const int base = t * CH;
  int s = 0;
  for (int i = 0; i < CH; ++i) { int idx = base + i; if (idx < n) s += deg[idx]; }
  part[t] = s;
  __syncthreads();
  for (int off = 1; off < 1024; off <<= 1) {
    int v = (t >= off) ? part[t - off] : 0;
    __syncthreads();
    part[t] += v;
    __syncthreads();
  }
  int run = (t == 0) ? 0 : part[t - 1];
  for (int i = 0; i < CH; ++i) {
    int idx = base + i;
    if (idx < n) { row_start[idx] = run; run += deg[idx]; }
  }
  if (t == 1023) row_start[n] = run;
}

__global__ void csr_fill_k(const int* __restrict__ dst, const int* __restrict__ row_start,
                           int* __restrict__ cursor, int* __restrict__ edge_by_dst, int E) {
  int e = blockIdx.x * blockDim.x + threadIdx.x;
  if (e < E) {
    int d = dst[e];
    int p = atomicAdd(&cursor[d], 1);
    edge_by_dst[row_start[d] + p] = e;
  }
}

// ------------------------- per-edge attention logits ------------------------
// One wave per edge; 8 lanes per head; shuffle-reduce the per-head dot.
template <int H, int C>
__global__ void edge_logits_k(const float* __restrict__ Q, const float* __restrict__ Km,
                              const int* __restrict__ src, const int* __restrict__ dst,
                              float* __restrict__ logits, unsigned* __restrict__ segkey,
                              int E, float scale) {
  const int HC = H * C;
  const int CPL = C / 8;
  int wid = blockIdx.x * (blockDim.x >> 5) + (threadIdx.x >> 5);
  if (wid >= E) return;
  const int lane = threadIdx.x & 31;
  const int h = lane >> 3;
  const int j = lane & 7;
  const int s = src[wid], d = dst[wid];
  const float* qr = Q + (size_t)d * HC + h * C + j * CPL;
  const float* kr = Km + (size_t)s * HC + h * C + j * CPL;
  float acc = 0.f;
#pragma unroll
  for (int i = 0; i < CPL; ++i) acc += qr[i] * kr[i];
  acc += __shfl_xor(acc, 1, 32);
  acc += __shfl_xor(acc, 2, 32);
  acc += __shfl_xor(acc, 4, 32);
  if (j == 0) {
    float lg = acc * scale;
    logits[(size_t)wid * H + h] = lg;
    atomicMax(&segkey[(size_t)d * H + h], fkey(lg));
  }
}

// ------------------------- segment softmax (in-place) -----------------------
// Thread per (node, head): exp(logit - max) written back over logits; 1/denom out.
__global__ void softmax_seg_k(float* __restrict__ logits, const unsigned* __restrict__ segkey,
                              const int* __restrict__ row_start, const int* __restrict__ edge_by_dst,
                              float* __restrict__ dinv, int Nn, int H) {
  int t = blockIdx.x * blockDim.x + threadIdx.x;
  if (t >= Nn * H) return;
  const int n = t / H;
  const int h = t - n * H;
  const int b = row_start[n], e1 = row_start[n + 1];
  const float rmax = funkey(segkey[t]);
  float s = 0.f;
  for (int i = b; i < e1; ++i) {
    const int e = edge_by_dst[i];
    const float ex = expf(logits[(size_t)e * H + h] - rmax);
    logits[(size_t)e * H + h] = ex;
    s += ex;
  }
  dinv[t] = (s > 0.f) ? 1.f / s : 0.f;
}

// ------------------------- weighted aggregation + skip (+ELU / head-mean) ---
// One wave per node; lanes own HC/32 channels (head-aligned).
template <int H, int C, bool MEAN>
__global__ void aggregate_k(const float* __restrict__ V, const float* __restrict__ S,
                            const float* __restrict__ alpha, const float* __restrict__ dinv,
                            const int* __restrict__ row_start, const int* __restrict__ edge_by_dst,
                            const int* __restrict__ src, float* __restrict__ Out, int Nn) {
  const int HC = H * C;
  const int CPL = HC / 32;
  int n = blockIdx.x * (blockDim.x >> 5) + (threadIdx.x >> 5);
  if (n >= Nn) return;
  const int lane = threadIdx.x & 31;
  const int h = lane >> 3;  // 8 lanes per head (32/H, H=4)
  float acc[CPL];
#pragma unroll
  for (int c = 0; c < CPL; ++c) acc[c] = 0.f;
  const float di = dinv[(size_t)n * H + h];
  const int b = row_start[n], e1 = row_start[n + 1];
  for (int i = b; i < e1; ++i) {
    const int e = edge_by_dst[i];
    const float w = alpha[(size_t)e * H + h] * di;
    const float* vr = V + (size_t)src[e] * HC + lane * CPL;
#pragma unroll
    for (int c = 0; c < CPL; ++c) acc[c] += w * vr[c];
  }
  if (!MEAN) {
    const float* sr = S + (size_t)n * HC;
    float* orow = Out + (size_t)n * HC;
#pragma unroll
    for (int c = 0; c < CPL; ++c) {
      const int ch = lane * CPL + c;
      const float x = acc[c] + sr[ch];
      orow[ch] = (x > 0.f) ? x : expm1f(x);  // fused ELU (inter-layer)
    }
  } else {
    // mean over heads: flat channel = h*C + c; lanes {l, l^8, l^16, l^24} share c
#pragma unroll
    for (int c = 0; c < CPL; ++c) {
      acc[c] += __shfl_xor(acc[c], 8, 32);
      acc[c] += __shfl_xor(acc[c], 16, 32);
    }
    if (lane < 8) {
      const float* sr = S + (size_t)n * C;
      float* orow = Out + (size_t)n * C;
#pragma unroll
      for (int c = 0; c < CPL; ++c) {
        const int ch = lane * CPL + c;  // 0..C-1
        orow[ch] = acc[c] * (1.f / H) + sr[ch];
      }
    }
  }
}

// ------------------------- host orchestration --------------------------------
static inline size_t align_up(size_t x, size_t a) { return (x + a - 1) & ~(a - 1); }

extern "C" void kernel_launch(void* const* d_in, const int* in_sizes, int n_in,
                              void* d_out, int out_size, void* d_ws, size_t ws_size,
                              hipStream_t stream) {
  (void)in_sizes; (void)n_in; (void)out_size; (void)ws_size;
  const int N = NNODES, E = NEDGES, H = NHEADS;

  const float* x = (const float*)d_in[0];
  const int* edge_index = (const int*)d_in[1];
  const int* srcv = edge_index;       // row 0
  const int* dstv = edge_index + E;   // row 1
  // params flat order per layer: Wq, Wk, Wv, Ws, bq, bk, bv, bs (dict insertion order)
  const float* P[3][8];
  for (int l = 0; l < 3; ++l)
    for (int j = 0; j < 8; ++j) P[l][j] = (const float*)d_in[3 + l * 8 + j];

  // ---- workspace carve-up ----
  char* w = (char*)d_ws;
  size_t off = 0;
  auto carve = [&](size_t bytes) { void* p = w + off; off = align_up(off + bytes, 256); return p; };
  float* qb   = (float*)carve((size_t)N * 256 * 4);
  float* kb   = (float*)carve((size_t)N * 256 * 4);
  float* vb   = (float*)carve((size_t)N * 256 * 4);
  float* sb   = (float*)carve((size_t)N * 256 * 4);
  float* hA   = (float*)carve((size_t)N * 256 * 4);
  float* hB   = (float*)carve((size_t)N * 256 * 4);
  float* lg   = (float*)carve((size_t)E * H * 4);   // logits, reused in-place as alpha
  float* dinv = (float*)carve((size_t)N * H * 4);
  unsigned* segkey = (unsigned*)carve((size_t)N * H * 4);
  int* deg       = (int*)carve((size_t)N * 4);
  int* row_start = (int*)carve((size_t)(N + 1) * 4);
  int* cursor    = (int*)carve((size_t)N * 4);
  int* edge_by_dst = (int*)carve((size_t)E * 4);

  const dim3 blk256(256);
  const int eblk = (E + 255) / 256;

  // ---- CSR-by-dst (built fresh every call; deterministic structure) ----
  hipMemsetAsync(deg, 0, (size_t)N * 4, stream);
  deg_count_k<<<eblk, blk256, 0, stream>>>(dstv, deg, E);
  scan_k<<<1, 1024, 0, stream>>>(deg, row_start, N);
  hipMemsetAsync(cursor, 0, (size_t)N * 4, stream);
  csr_fill_k<<<eblk, blk256, 0, stream>>>(dstv, row_start, cursor, edge_by_dst, E);

  const int Mtiles = N / 16;  // 625
  const int wavesPerBlk = 256 / 32;
  const int edgeBlocks = (E + wavesPerBlk - 1) / wavesPerBlk;
  const int nodeBlocks = (N + wavesPerBlk - 1) / wavesPerBlk;
  const int nhBlocks = (N * H + 255) / 256;

  // ================= layer 0: fin=500, C=64, HC=256, concat+ELU =============
  {
    const int K = 500;
    dim3 g(Mtiles, 4);  // 256/64
    gemm_bias_wmma<256><<<g, 32, 0, stream>>>(x, P[0][0], P[0][4], qb, K);
    gemm_bias_wmma<256><<<g, 32, 0, stream>>>(x, P[0][1], P[0][5], kb, K);
    gemm_bias_wmma<256><<<g, 32, 0, stream>>>(x, P[0][2], P[0][6], vb, K);
    gemm_bias_wmma<256><<<g, 32, 0, stream>>>(x, P[0][3], P[0][7], sb, K);
    hipMemsetAsync(segkey, 0, (size_t)N * H * 4, stream);
    edge_logits_k<4, 64><<<edgeBlocks, blk256, 0, stream>>>(qb, kb, srcv, dstv, lg, segkey, E, 0.125f);
    softmax_seg_k<<<nhBlocks, blk256, 0, stream>>>(lg, segkey, row_start, edge_by_dst, dinv, N, H);
    aggregate_k<4, 64, false><<<nodeBlocks, blk256, 0, stream>>>(vb, sb, lg, dinv, row_start, edge_by_dst, srcv, hA, N);
  }

  // ================= layer 1: fin=256, C=64, HC=256, concat+ELU =============
  {
    const int K = 256;
    dim3 g(Mtiles, 4);
    gemm_bias_wmma<256><<<g, 32, 0, stream>>>(hA, P[1][0], P[1][4], qb, K);
    gemm_bias_wmma<256><<<g, 32, 0, stream>>>(hA, P[1][1], P[1][5], kb, K);
    gemm_bias_wmma<256><<<g, 32, 0, stream>>>(hA, P[1][2], P[1][6], vb, K);
    gemm_bias_wmma<256><<<g, 32, 0, stream>>>(hA, P[1][3], P[1][7], sb, K);
    hipMemsetAsync(segkey, 0, (size_t)N * H * 4, stream);
    edge_logits_k<4, 64><<<edgeBlocks, blk256, 0, stream>>>(qb, kb, srcv, dstv, lg, segkey, E, 0.125f);
    softmax_seg_k<<<nhBlocks, blk256, 0, stream>>>(lg, segkey, row_start, edge_by_dst, dinv, N, H);
    aggregate_k<4, 64, false><<<nodeBlocks, blk256, 0, stream>>>(vb, sb, lg, dinv, row_start, edge_by_dst, srcv, hB, N);
  }

  // ================= layer 2: fin=256, C=40, HC=160, head-mean, no ELU ======
  {
    const int K = 256;
    dim3 g(Mtiles, 3);   // ceil(160/64)
    gemm_bias_wmma<160><<<g, 32, 0, stream>>>(hB, P[2][0], P[2][4], qb, K);
    gemm_bias_wmma<160><<<g, 32, 0, stream>>>(hB, P[2][1], P[2][5], kb, K);
    gemm_bias_wmma<160><<<g, 32, 0, stream>>>(hB, P[2][2], P[2][6], vb, K);
    dim3 gs(Mtiles, 1);  // ceil(40/64)
    gemm_bias_wmma<40><<<gs, 32, 0, stream>>>(hB, P[2][3], P[2][7], sb, K);
    hipMemsetAsync(segkey, 0, (size_t)N * H * 4, stream);
    const float scale2 = 1.0f / sqrtf(40.0f);
    edge_logits_k<4, 40><<<edgeBlocks, blk256, 0, stream>>>(qb, kb, srcv, dstv, lg, segkey, E, scale2);
    softmax_seg_k<<<nhBlocks, blk256, 0, stream>>>(lg, segkey, row_start, edge_by_dst, dinv, N, H);
    aggregate_k<4, 40, true><<<nodeBlocks, blk256, 0, stream>>>(vb, sb, lg, dinv, row_start, edge_by_dst, srcv, (float*)d_out, N);
  }
}